// AlphaGridMask_90202903151142
// MI455X (gfx1250) — compile-verified
//
#include <hip/hip_runtime.h>
#include <stdint.h>

// ---------------------------------------------------------------------------
// AlphaGridMask (KiloNeRF) for MI455X / gfx1250, wave32.
//  Pass A: per-CTA 64-bin histograms of block ids            (counts[bin][cta])
//  Pass B: exclusive scan of bin-major counts                (stable sort bases)
//  Pass C: async-LDS staged stream; stable wave32-ballot rank; trilinear
//          sample of the 64x(64^3) alpha volume; scatter to sorted position.
// ---------------------------------------------------------------------------

#define NBINS        64
#define CTA_THREADS  256
#define CHUNKS       16
#define PTS_PER_CTA  (CTA_THREADS * CHUNKS)   // 4096

typedef __attribute__((address_space(1))) int as1_int;
typedef __attribute__((address_space(3))) int as3_int;

// ---- CDNA5 async global->LDS copy (ASYNCcnt path) -------------------------
__device__ __forceinline__ void async_g2l_b32(const float* gsrc, float* lds_dst) {
#if __has_builtin(__builtin_amdgcn_global_load_async_to_lds_b32)
  __builtin_amdgcn_global_load_async_to_lds_b32(
      (as1_int*)(uintptr_t)gsrc,
      (as3_int*)(uint32_t)(uintptr_t)lds_dst,
      0 /*offset*/, 0 /*cpol*/);
#else
  unsigned lds_off = (unsigned)(uintptr_t)lds_dst;  // low 32 bits of flat LDS addr == LDS byte offset
  asm volatile("global_load_async_to_lds_b32 %0, %1, off"
               :: "v"(lds_off), "v"(gsrc) : "memory");
#endif
}

#if __has_builtin(__builtin_amdgcn_s_wait_asynccnt)
#define WAIT_ASYNC(n) __builtin_amdgcn_s_wait_asynccnt(n)
#else
#define WAIT_ASYNC(n) asm volatile("s_wait_asynccnt %0" :: "n"(n) : "memory")
#endif

// ---- wave32 ballot --------------------------------------------------------
__device__ __forceinline__ unsigned ballot32(bool p) {
#if __has_builtin(__builtin_amdgcn_ballot_w32)
  return __builtin_amdgcn_ballot_w32(p);
#else
  return (unsigned)__ballot(p);
#endif
}

// ---- block id, identical arithmetic to the reference ----------------------
__device__ __forceinline__ int compute_bid(float x, float y, float z,
                                           const float* __restrict__ aabb) {
  const float a0x = aabb[0], a0y = aabb[1], a0z = aabb[2];
  const float vx = (aabb[3] - a0x) / 4.0f;
  const float vy = (aabb[4] - a0y) / 4.0f;
  const float vz = (aabb[5] - a0z) / 4.0f;
  int ix = (int)floorf((x - a0x) / vx);
  int iy = (int)floorf((y - a0y) / vy);
  int iz = (int)floorf((z - a0z) / vz);
  ix = ix < 0 ? 0 : (ix > 3 ? 3 : ix);
  iy = iy < 0 ? 0 : (iy > 3 ? 3 : iy);
  iz = iz < 0 ? 0 : (iz > 3 ? 3 : iz);
  return ix * 16 + iy * 4 + iz;
}

// ---------------------------------------------------------------------------
// Pass A: per-CTA histogram -> counts[bin*numCTA + cta]
// ---------------------------------------------------------------------------
__global__ void agm_count_kernel(const float* __restrict__ xyz,
                                 const float* __restrict__ aabb,
                                 unsigned* __restrict__ counts,
                                 int n, int numCTA) {
  __shared__ unsigned hist[NBINS];
  const int tid = threadIdx.x;
  const int cta = blockIdx.x;
  if (tid < NBINS) hist[tid] = 0u;
  __syncthreads();

  const long base = (long)cta * PTS_PER_CTA;
  for (int j = 0; j < CHUNKS; ++j) {
    const long i = base + (long)j * CTA_THREADS + tid;
    if (i < n) {
      const float x = xyz[i * 3 + 0];
      const float y = xyz[i * 3 + 1];
      const float z = xyz[i * 3 + 2];
      atomicAdd(&hist[compute_bid(x, y, z, aabb)], 1u);
    }
  }
  __syncthreads();
  if (tid < NBINS) counts[(long)tid * numCTA + cta] = hist[tid];
}

// ---------------------------------------------------------------------------
// Pass B: single-workgroup exclusive scan of M = NBINS*numCTA entries
// ---------------------------------------------------------------------------
__global__ void agm_scan_kernel(unsigned* __restrict__ data, int M) {
  __shared__ unsigned tot[1024];
  const int tid = threadIdx.x;
  const int T = blockDim.x;
  const int seg = (M + T - 1) / T;
  const long s = (long)tid * seg;

  unsigned run = 0;
  for (int e = 0; e < seg; ++e) {
    const long idx = s + e;
    if (idx < M) { const unsigned v = data[idx]; data[idx] = run; run += v; }
  }
  tot[tid] = run;
  __syncthreads();
  for (int off = 1; off < T; off <<= 1) {
    const unsigned v = (tid >= off) ? tot[tid - off] : 0u;
    __syncthreads();
    tot[tid] += v;
    __syncthreads();
  }
  const unsigned basev = tid ? tot[tid - 1] : 0u;
  for (int e = 0; e < seg; ++e) {
    const long idx = s + e;
    if (idx < M) data[idx] += basev;
  }
}

// ---------------------------------------------------------------------------
// Pass C: async-LDS staging, stable rank via ballot match, trilinear sample,
//         scatter to sorted output position.
// ---------------------------------------------------------------------------
__global__ void agm_sample_scatter_kernel(const float* __restrict__ xyz,
                                          const float* __restrict__ aabb,
                                          const float* __restrict__ vol,
                                          const float* __restrict__ dmin,
                                          const float* __restrict__ dmax,
                                          const unsigned* __restrict__ bases,
                                          float* __restrict__ out,
                                          int n, int numCTA) {
  __shared__ float    sxyz[2][CTA_THREADS * 3];
  __shared__ unsigned wh[8][NBINS];     // per-wave per-bin counts of a chunk
  __shared__ unsigned running[NBINS];   // counts consumed in earlier chunks
  __shared__ unsigned cbase[NBINS];     // scanned global base for (bin, cta)

  const int tid  = threadIdx.x;
  const int cta  = blockIdx.x;
  const unsigned lane = tid & 31u;      // gfx1250 is wave32
  const unsigned wv   = tid >> 5;
  unsigned* whf = &wh[0][0];

  if (tid < NBINS) {
    running[tid] = 0u;
    cbase[tid]   = bases[(long)tid * numCTA + cta];
  }

  const long base = (long)cta * PTS_PER_CTA;

  // Prefetch chunk 0 into LDS buffer 0 (clamp tail addresses; values unused).
  {
    long i = base + tid;
    if (i >= n) i = n - 1;
    const float* src = xyz + i * 3;
    async_g2l_b32(src + 0, &sxyz[0][tid * 3 + 0]);
    async_g2l_b32(src + 1, &sxyz[0][tid * 3 + 1]);
    async_g2l_b32(src + 2, &sxyz[0][tid * 3 + 2]);
  }

  for (int j = 0; j < CHUNKS; ++j) {
    const int buf = j & 1;

    // zero this chunk's per-wave histograms (512 entries / 256 threads)
    whf[tid] = 0u;
    whf[tid + 256] = 0u;

    // double-buffered async staging: kick next chunk, wait for current
    if (j + 1 < CHUNKS) {
      long i2 = base + (long)(j + 1) * CTA_THREADS + tid;
      if (i2 >= n) i2 = n - 1;
      const float* src = xyz + i2 * 3;
      async_g2l_b32(src + 0, &sxyz[buf ^ 1][tid * 3 + 0]);
      async_g2l_b32(src + 1, &sxyz[buf ^ 1][tid * 3 + 1]);
      async_g2l_b32(src + 2, &sxyz[buf ^ 1][tid * 3 + 2]);
      WAIT_ASYNC(3);   // oldest 3 (current chunk) retired
    } else {
      WAIT_ASYNC(0);
    }
    __syncthreads();   // wh zeroed + running update from prev chunk visible

    const long i = base + (long)j * CTA_THREADS + tid;
    const bool valid = (i < n);
    const float x = sxyz[buf][tid * 3 + 0];
    const float y = sxyz[buf][tid * 3 + 1];
    const float z = sxyz[buf][tid * 3 + 2];
    const int bid = compute_bid(x, y, z, aabb);

    // stable intra-wave match: lanes with equal bid, restricted to valid lanes
    unsigned m = ballot32(valid);
    #pragma unroll
    for (int k = 0; k < 6; ++k) {
      const unsigned b = ballot32(((unsigned)bid >> k) & 1u);
      m &= (((unsigned)bid >> k) & 1u) ? b : ~b;
    }
    const unsigned rank = __popc(m & ((1u << lane) - 1u));
    const unsigned cnt  = __popc(m);
    if (valid && rank == 0u) wh[wv][bid] = cnt;
    __syncthreads();   // wh complete for this chunk

    if (valid) {
      unsigned off = running[bid] + rank;
      for (unsigned w = 0; w < wv; ++w) off += wh[w][bid];
      const unsigned pos = cbase[bid] + off;

      // global_to_local + grid_sample(align_corners=True, zeros padding)
      const float dn0 = dmin[bid * 3 + 0], dn1 = dmin[bid * 3 + 1], dn2 = dmin[bid * 3 + 2];
      const float dx0 = dmax[bid * 3 + 0], dx1 = dmax[bid * 3 + 1], dx2 = dmax[bid * 3 + 2];
      const float lx = 2.0f * (x - dn0) / (dx0 - dn0) - 1.0f;
      const float ly = 2.0f * (y - dn1) / (dx1 - dn1) - 1.0f;
      const float lz = 2.0f * (z - dn2) / (dx2 - dn2) - 1.0f;
      const float fx = (lx + 1.0f) * 0.5f * 63.0f;
      const float fy = (ly + 1.0f) * 0.5f * 63.0f;
      const float fz = (lz + 1.0f) * 0.5f * 63.0f;
      const float x0f = floorf(fx), y0f = floorf(fy), z0f = floorf(fz);
      const float wx = fx - x0f, wy = fy - y0f, wz = fz - z0f;
      const int x0 = (int)x0f, y0 = (int)y0f, z0 = (int)z0f;

      const float* blk = vol + ((long)bid << 18);   // 64^3 per block
      float acc = 0.0f;
      #pragma unroll
      for (int dz = 0; dz < 2; ++dz) {
        const float wz_ = dz ? wz : (1.0f - wz);
        const int zi = z0 + dz;
        #pragma unroll
        for (int dy = 0; dy < 2; ++dy) {
          const float wy_ = dy ? wy : (1.0f - wy);
          const int yi = y0 + dy;
          #pragma unroll
          for (int dx = 0; dx < 2; ++dx) {
            const float wx_ = dx ? wx : (1.0f - wx);
            const int xi = x0 + dx;
            const bool inb = (zi >= 0) & (zi < 64) & (yi >= 0) & (yi < 64) &
                             (xi >= 0) & (xi < 64);
            const int zc = zi < 0 ? 0 : (zi > 63 ? 63 : zi);
            const int yc = yi < 0 ? 0 : (yi > 63 ? 63 : yi);
            const int xc = xi < 0 ? 0 : (xi > 63 ? 63 : xi);
            const float v = blk[(zc << 12) + (yc << 6) + xc];
            acc = acc + wz_ * wy_ * wx_ * (inb ? v : 0.0f);
          }
        }
      }
      out[pos] = acc;
    }
    __syncthreads();   // all readers of wh/running done

    if (tid < NBINS) {
      unsigned s = 0;
      #pragma unroll
      for (int w = 0; w < 8; ++w) s += wh[w][tid];
      running[tid] += s;
    }
    __syncthreads();   // protect wh from next chunk's zeroing
  }
}

// ---------------------------------------------------------------------------
extern "C" void kernel_launch(void* const* d_in, const int* in_sizes, int n_in,
                              void* d_out, int out_size, void* d_ws, size_t ws_size,
                              hipStream_t stream) {
  const float* xyz  = (const float*)d_in[0];  // [N,3]
  const float* aabb = (const float*)d_in[1];  // [2,3]
  const float* vol  = (const float*)d_in[2];  // [256,256,256] == [64,64,64,64]
  const float* dmin = (const float*)d_in[3];  // [64,3]
  const float* dmax = (const float*)d_in[4];  // [64,3]
  float* out = (float*)d_out;

  const int n = in_sizes[0] / 3;
  const int numCTA = (n + PTS_PER_CTA - 1) / PTS_PER_CTA;   // 1024 for N=4M
  unsigned* counts = (unsigned*)d_ws;                       // NBINS*numCTA u32 (256 KB)
  const int M = NBINS * numCTA;

  agm_count_kernel<<<numCTA, CTA_THREADS, 0, stream>>>(xyz, aabb, counts, n, numCTA);
  agm_scan_kernel<<<1, 1024, 0, stream>>>(counts, M);
  agm_sample_scatter_kernel<<<numCTA, CTA_THREADS, 0, stream>>>(
      xyz, aabb, vol, dmin, dmax, counts, out, n, numCTA);
}